// Rule_67662914781184
// MI455X (gfx1250) — compile-verified
//
#include <hip/hip_runtime.h>
#include <cstdint>

// Ising/Metropolis step, 2048x2048 periodic grid, B=2, C=1.
// Memory-bound (AI ~0.4 flop/byte): optimize purely for data movement.
//  - one block per row (512 thr x 4 px): x-wrap is free inside the block
//  - s rows (3 per batch) staged to LDS via gfx1250 async-to-LDS copies
//  - J / rand_drop read ONCE and applied to both batches (J is the big stream)
//  - all single-use streams use non-temporal b128 accesses

#define HH 2048
#define WW 2048
#define NPIX (HH * WW)
#define NBATCH 2
#define DECAY 0.95f

typedef __attribute__((ext_vector_type(4))) float f4;
typedef __attribute__((ext_vector_type(4))) int   v4i;

#define AS1 __attribute__((address_space(1)))
#define AS3 __attribute__((address_space(3)))

__device__ __forceinline__ void async_copy_f4(const float* g, float* l) {
#if __has_builtin(__builtin_amdgcn_global_load_async_to_lds_b128)
  // signature (from clang diagnostic): (v4i AS1*, v4i AS3*, imm offset, imm cpol)
  __builtin_amdgcn_global_load_async_to_lds_b128((AS1 v4i*)g, (AS3 v4i*)l, 0, 0);
#else
  unsigned lds_off = (unsigned)(uintptr_t)(AS3 float*)l;
  asm volatile("global_load_async_to_lds_b128 %0, %1, off"
               :: "v"(lds_off), "v"(g) : "memory");
#endif
}

__device__ __forceinline__ void wait_async_all() {
#if __has_builtin(__builtin_amdgcn_s_wait_asynccnt)
  __builtin_amdgcn_s_wait_asynccnt(0);
#else
  asm volatile("s_wait_asynccnt 0" ::: "memory");
#endif
}

__global__ __launch_bounds__(512)
void ising_metropolis(const float* __restrict__ xin,
                      const float* __restrict__ obvs,
                      const float* __restrict__ Jm,
                      const float* __restrict__ rand_s,
                      const float* __restrict__ rand_drop,
                      float* __restrict__ out)
{
  // tile[batch][row(-1,0,+1)][x] : 2*3*2048 floats = 48 KB of the 320 KB WGP LDS
  __shared__ float tile[NBATCH][3][WW];

  const int y   = (int)blockIdx.x;
  const int t   = (int)threadIdx.x;
  const int x0  = t << 2;                       // 4 pixels per thread
  const int ym1 = (y + HH - 1) & (HH - 1);
  const int yp1 = (y + 1) & (HH - 1);
  const size_t rowOff = (size_t)y * WW + x0;

  // ---- stage s rows for both batches into LDS (async, ASYNCcnt-tracked) ----
#pragma unroll
  for (int bb = 0; bb < NBATCH; ++bb) {
    const float* sP = xin + (size_t)bb * 2 * NPIX;   // s plane of batch bb
    async_copy_f4(sP + (size_t)ym1 * WW + x0, &tile[bb][0][x0]);
    async_copy_f4(sP + (size_t)y   * WW + x0, &tile[bb][1][x0]);
    async_copy_f4(sP + (size_t)yp1 * WW + x0, &tile[bb][2][x0]);
  }

  // ---- overlap: stream in batch-shared data while async copies fly ----
  f4 Jv[9];
#pragma unroll
  for (int k = 0; k < 9; ++k)
    Jv[k] = __builtin_nontemporal_load((const f4*)(Jm + (size_t)k * NPIX + rowOff));
  const f4 rdrop = __builtin_nontemporal_load((const f4*)(rand_drop + rowOff));

  wait_async_all();
  __syncthreads();

  const int xl = (x0 - 1) & (WW - 1);   // periodic wrap inside the row
  const int xr = (x0 + 4) & (WW - 1);

#pragma unroll
  for (int bb = 0; bb < NBATCH; ++bb) {
    // gather stencil values from LDS: 4 centers + 1 left + 1 right per row
    f4 C[3]; float L[3], R[3];
#pragma unroll
    for (int r = 0; r < 3; ++r) {
      C[r] = *(const f4*)&tile[bb][r][x0];
      L[r] = tile[bb][r][xl];
      R[r] = tile[bb][r][xr];
    }

    // acc = sum_k J_k * neighbor_k   (k = row*3 + col, row0 = y-1, col0 = x-1)
    f4 acc;
#pragma unroll
    for (int p = 0; p < 4; ++p) {
      float a = 0.0f;
#pragma unroll
      for (int r = 0; r < 3; ++r) {
        const float ln = (p == 0) ? L[r] : C[r][p - 1];
        const float cn = C[r][p];
        const float rn = (p == 3) ? R[r] : C[r][p + 1];
        a = fmaf(Jv[3 * r + 0][p], ln, a);
        a = fmaf(Jv[3 * r + 1][p], cn, a);   // center J is zero in input
        a = fmaf(Jv[3 * r + 2][p], rn, a);
      }
      acc[p] = a;
    }

    const f4 s  = C[1];
    const f4 de = 2.0f * s * acc;       // delta_e
    const f4 E  = -0.5f * de;

    const size_t xb = (size_t)bb * 2 * NPIX;
    const size_t ob = (size_t)bb * 4 * NPIX;
    const f4 bf = __builtin_nontemporal_load((const f4*)(xin + xb + (size_t)NPIX + rowOff));
    f4 e  = __builtin_nontemporal_load((const f4*)(obvs + ob + 0 * (size_t)NPIX + rowOff));
    f4 e2 = __builtin_nontemporal_load((const f4*)(obvs + ob + 1 * (size_t)NPIX + rowOff));
    f4 cv = __builtin_nontemporal_load((const f4*)(obvs + ob + 2 * (size_t)NPIX + rowOff));
    f4 m  = __builtin_nontemporal_load((const f4*)(obvs + ob + 3 * (size_t)NPIX + rowOff));
    const f4 rs = __builtin_nontemporal_load((const f4*)(rand_s + (size_t)bb * NPIX + rowOff));

    const float d = DECAY, omd = 1.0f - DECAY;
    e  = d * e  + omd * E;
    e2 = d * e2 + omd * (E * E);
    cv = d * cv + omd * (e2 - e * e) * (bf * bf);
    m  = d * m  + omd * s;

    f4 sout;
#pragma unroll
    for (int p = 0; p < 4; ++p) {
      const float pa = (de[p] <= 0.0f) ? 1.0f : __expf(-de[p] * bf[p]);
      const bool accept = (rs[p] < pa) && (rdrop[p] > 0.5f);
      sout[p] = accept ? -s[p] : s[p];
    }

    // outputs: state (B,2,H,W) then obvs_out (B,4,H,W), concatenated flat
    float* st = out + xb;
    __builtin_nontemporal_store(sout, (f4*)(st + rowOff));
    __builtin_nontemporal_store(bf,   (f4*)(st + (size_t)NPIX + rowOff));
    float* ov = out + (size_t)4 * NPIX + ob;
    __builtin_nontemporal_store(e,  (f4*)(ov + 0 * (size_t)NPIX + rowOff));
    __builtin_nontemporal_store(e2, (f4*)(ov + 1 * (size_t)NPIX + rowOff));
    __builtin_nontemporal_store(cv, (f4*)(ov + 2 * (size_t)NPIX + rowOff));
    __builtin_nontemporal_store(m,  (f4*)(ov + 3 * (size_t)NPIX + rowOff));
  }
}

extern "C" void kernel_launch(void* const* d_in, const int* in_sizes, int n_in,
                              void* d_out, int out_size, void* d_ws, size_t ws_size,
                              hipStream_t stream) {
  (void)in_sizes; (void)n_in; (void)out_size; (void)d_ws; (void)ws_size;
  const float* xin       = (const float*)d_in[0];  // (B, 2, H, W)
  const float* obvs      = (const float*)d_in[1];  // (B, 4, H, W)
  const float* Jm        = (const float*)d_in[2];  // (1, 9, N)
  const float* rand_s    = (const float*)d_in[3];  // (B, 1, H, W)
  const float* rand_drop = (const float*)d_in[4];  // (1, 1, H, W)
  float* out = (float*)d_out;                      // state (4N) ++ obvs (8N)

  ising_metropolis<<<dim3(HH), dim3(512), 0, stream>>>(
      xin, obvs, Jm, rand_s, rand_drop, out);
}